// Reconstructor_8461085573440
// MI455X (gfx1250) — compile-verified
//
#include <hip/hip_runtime.h>
#include <stdint.h>

// Problem constants (from reference)
#define NUM_LUT 3
#define IN_F    4096
#define OUT_F   4096
#define VEC     4
#define LUTSZ   16
#define NGROUP  32            // IN_F / 128 quantization groups
#define G_TOT   1024          // IN_F / VEC vector-groups

// Tiling
#define O_TILE  32
#define G_TILE  32
#define THREADS 256
#define WROW    (G_TILE * VEC + 4)   // 132 floats: padded row to dodge LDS bank conflicts

#if defined(__AMDGCN__) && __has_builtin(__builtin_amdgcn_global_load_async_to_lds_b128) && __has_builtin(__builtin_amdgcn_s_wait_asynccnt)
#define USE_ASYNC_LDS 1
#else
#define USE_ASYNC_LDS 0
#endif

#if USE_ASYNC_LDS
typedef int async_v4i __attribute__((vector_size(16)));
typedef __attribute__((address_space(1))) async_v4i* glob_v4i_ptr;
typedef __attribute__((address_space(3))) async_v4i* lds_v4i_ptr;
#endif

__device__ __forceinline__ void argmax4(float4 f, int sbase, float& best, int& bi) {
  if (f.x > best) { best = f.x; bi = sbase + 0; }
  if (f.y > best) { best = f.y; bi = sbase + 1; }
  if (f.z > best) { best = f.z; bi = sbase + 2; }
  if (f.w > best) { best = f.w; bi = sbase + 3; }
}

__global__ __launch_bounds__(THREADS) void recon_kernel(
    const float* __restrict__ gate,       // [L, G, O, S]
    const float* __restrict__ codebook,   // [L, G, S, V]
    const float* __restrict__ scales,     // [O, NG]
    const int*   __restrict__ zeros,      // [O, NG]
    float* __restrict__ out)              // [O, IN_F]
{
  __shared__ __align__(16) float cb[NUM_LUT * G_TILE * LUTSZ * VEC]; // 6144 floats = 24 KB
  __shared__ __align__(16) float wt[O_TILE * WROW];                  // ~16.5 KB

  const int t     = threadIdx.x;
  const int oBase = blockIdx.x * O_TILE;
  const int gTile = blockIdx.y;            // == quantization group index ng
  const int gBase = gTile * G_TILE;

  // ---------------- Stage codebook tile (3 contiguous 8 KB slabs) -> LDS ----------------
  {
    const int F4_TOTAL = NUM_LUT * G_TILE * LUTSZ;   // 1536 float4 chunks
    for (int i = t; i < F4_TOTAL; i += THREADS) {
      const int l = i / (G_TILE * LUTSZ);
      const int r = i - l * (G_TILE * LUTSZ);
      const float* src = codebook + (size_t)(l * G_TOT + gBase) * (LUTSZ * VEC) + (size_t)r * 4;
      float* dst = &cb[l * (G_TILE * LUTSZ * VEC) + r * 4];
#if USE_ASYNC_LDS
      __builtin_amdgcn_global_load_async_to_lds_b128(
          (glob_v4i_ptr)src, (lds_v4i_ptr)dst, 0, 0);
#else
      *(float4*)dst = *(const float4*)src;
#endif
    }
#if USE_ASYNC_LDS
    __builtin_amdgcn_s_wait_asynccnt(0);
#endif
  }
  __syncthreads();

  // ---------------- Phase 1: stream gate, argmax+gather, accumulate over LUTs ----------------
  const int oLocal = t & (O_TILE - 1);   // lanes take consecutive o -> coalesced 64B/lane
  const int wv     = t >> 5;             // wave id 0..7 selects g within tile
  const int oAbs   = oBase + oLocal;

  #pragma unroll
  for (int j = 0; j < 4; ++j) {
    const int gLocal = wv + j * 8;
    const int gAbs   = gBase + gLocal;

    if (j < 3) {  // prefetch next g-group's gate lines (global_prefetch_b8)
      const int gNext = gBase + wv + (j + 1) * 8;
      #pragma unroll
      for (int l = 0; l < NUM_LUT; ++l)
        __builtin_prefetch(gate + (((size_t)l * G_TOT + gNext) * OUT_F + oAbs) * LUTSZ, 0, 0);
    }

    // 12 independent global_load_b128 in flight (3 LUTs x 16 floats)
    float4 q[NUM_LUT][4];
    #pragma unroll
    for (int l = 0; l < NUM_LUT; ++l) {
      const float4* gp = (const float4*)(gate + (((size_t)l * G_TOT + gAbs) * OUT_F + oAbs) * LUTSZ);
      q[l][0] = gp[0]; q[l][1] = gp[1]; q[l][2] = gp[2]; q[l][3] = gp[3];
    }

    float ax = 0.f, ay = 0.f, az = 0.f, aw = 0.f;
    #pragma unroll
    for (int l = 0; l < NUM_LUT; ++l) {
      float best = q[l][0].x; int bi = 0;      // first-occurrence argmax (strict >)
      argmax4(q[l][0],  0, best, bi);
      argmax4(q[l][1],  4, best, bi);
      argmax4(q[l][2],  8, best, bi);
      argmax4(q[l][3], 12, best, bi);
      const float4 c = *(const float4*)&cb[(l * G_TILE + gLocal) * (LUTSZ * VEC) + bi * 4];
      ax += c.x; ay += c.y; az += c.z; aw += c.w;
    }
    *(float4*)&wt[oLocal * WROW + gLocal * 4] = make_float4(ax, ay, az, aw);
  }
  __syncthreads();

  // ---------------- Phase 2: dequantize + coalesced store (transpose via LDS) ----------------
  const int cc = t & 31;     // float4 chunk within 128-float output row == gLocal
  const int rB = t >> 5;
  #pragma unroll
  for (int j = 0; j < 4; ++j) {
    const int r  = rB + j * 8;
    const int oA = oBase + r;
    const float s = scales[(size_t)oA * NGROUP + gTile];       // wave-uniform -> scalar load
    const float z = (float)zeros[(size_t)oA * NGROUP + gTile];
    const float4 w = *(const float4*)&wt[r * WROW + cc * 4];
    float4 o4;
    o4.x = (w.x - z) * s;
    o4.y = (w.y - z) * s;
    o4.z = (w.z - z) * s;
    o4.w = (w.w - z) * s;
    *(float4*)&out[(size_t)oA * IN_F + gTile * (G_TILE * VEC) + cc * 4] = o4;
  }
}

extern "C" void kernel_launch(void* const* d_in, const int* in_sizes, int n_in,
                              void* d_out, int out_size, void* d_ws, size_t ws_size,
                              hipStream_t stream) {
  const float* gate     = (const float*)d_in[0];
  const float* codebook = (const float*)d_in[1];
  const float* scales   = (const float*)d_in[2];
  const int*   zeros    = (const int*)d_in[3];
  float* out            = (float*)d_out;
  (void)in_sizes; (void)n_in; (void)out_size; (void)d_ws; (void)ws_size;

  dim3 grid(OUT_F / O_TILE, G_TOT / G_TILE);   // 128 x 32 blocks
  recon_kernel<<<grid, dim3(THREADS), 0, stream>>>(gate, codebook, scales, zeros, out);
}